// ARIMA_86174223827540
// MI455X (gfx1250) — compile-verified
//
#include <hip/hip_runtime.h>
#include <hip/hip_bf16.h>

typedef __attribute__((ext_vector_type(2))) float v2f;
typedef __attribute__((ext_vector_type(8))) float v8f;

#define B_    256
#define T_    16384
#define F_    32
#define STEPS 16379           // (T-1) - max(p,q)
#define OUTW  16380           // STEPS + 1
#define NCH   64              // chunks for parallel recurrence
#define CHUNK 256             // power of two (M^CHUNK via 8 squarings)

// ---------------------------------------------------------------------------
// K1: base[t][b] = AR + feat + bias ; d[t][b] = tgt - base   (transposed layout)
// feat via v_wmma_f32_16x16x4_f32: A = 16 timesteps x 32 features (8 K=4 frags),
// B = weights replicated across all 16 columns -> every column of D = dot().
// ---------------------------------------------------------------------------
__global__ void __launch_bounds__(256)
k1_base(const float* __restrict__ x, const float* __restrict__ features,
        const float* __restrict__ ar_coef, const float* __restrict__ fw,
        const float* __restrict__ bias,
        float* __restrict__ baseT, float* __restrict__ dT) {
  const int wave = threadIdx.x >> 5;
  const int lane = threadIdx.x & 31;
  const int hi   = lane >> 4;          // half-wave selects K-pair (A/B) / M-half (D)
  const int mrow = lane & 15;          // A-matrix row (timestep within tile)
  const int b    = blockIdx.y;
  const int t0   = blockIdx.x * 128 + wave * 16;   // 16-timestep tile per wave

  int trow = t0 + 4 + mrow;            // features row = m + t, m = 4
  if (trow > T_ - 1) trow = T_ - 1;    // clamp tail (results guarded at store)
  const float* fp = features + ((size_t)b * T_ + trow) * F_ + hi * 2;

  v2f a[8];
#pragma unroll
  for (int kc = 0; kc < 8; ++kc)       // A frag: lane m holds K = hi*2 + {0,1}
    a[kc] = *(const v2f*)(fp + kc * 4);

  v2f bf[8];
#pragma unroll
  for (int kc = 0; kc < 8; ++kc) {     // B frag: same K indexing, all columns equal
    bf[kc].x = fw[kc * 4 + hi * 2 + 0];
    bf[kc].y = fw[kc * 4 + hi * 2 + 1];
  }

  v8f acc = {};
#pragma unroll
  for (int kc = 0; kc < 8; ++kc)
    acc = __builtin_amdgcn_wmma_f32_16x16x4_f32(
        false, a[kc], false, bf[kc], (short)0, acc, false, false);

  // D layout: lane 0 holds rows 0..7 in acc[0..7]; lane 16 holds rows 8..15.
  if (mrow == 0) {
    const float a0 = ar_coef[0], a1 = ar_coef[1], a2 = ar_coef[2], a3 = ar_coef[3];
    const float bv = bias[0];
    const float* xb = x + (size_t)b * T_;
#pragma unroll
    for (int i = 0; i < 8; ++i) {
      int t = t0 + hi * 8 + i;
      if (t < STEPS) {
        float x0 = xb[t], x1 = xb[t+1], x2 = xb[t+2], x3 = xb[t+3],
              x4 = xb[t+4], x5 = xb[t+5];
        float ar = a0*(x1-x0) + a1*(x2-x1) + a2*(x3-x2) + a3*(x4-x3);
        float basev = ar + acc[i] + bv;
        float tgt   = x5 - x4;                    // xd[:, m+t]
        baseT[(size_t)t * B_ + b] = basev;
        dT[(size_t)t * B_ + b]    = tgt - basev;  // d_t
      }
    }
  }
}

// ---------------------------------------------------------------------------
// K2: per (chunk c, series b): run recurrence from zero state -> E_c = (r1, r0)
// res_t = d_t - (t>=2 ? c1*r1 + c0*r0 : 0)
// ---------------------------------------------------------------------------
__global__ void __launch_bounds__(256)
k2_phase1(const float* __restrict__ dT, const float* __restrict__ ma_coef,
          float* __restrict__ E) {
  const int b = threadIdx.x, c = blockIdx.x;
  const float c0 = ma_coef[0], c1 = ma_coef[1];
  const int tb = c * CHUNK;
  const int te = (tb + CHUNK < STEPS) ? tb + CHUNK : STEPS;
  float r1 = 0.f, r0 = 0.f;
#pragma unroll 4
  for (int t = tb; t < te; ++t) {
    float mav = (t >= 2) ? (c1 * r1 + c0 * r0) : 0.f;
    float res = dT[(size_t)t * B_ + b] - mav;
    r0 = r1; r1 = res;
  }
  E[(c * B_ + b) * 2 + 0] = r1;
  E[(c * B_ + b) * 2 + 1] = r0;
}

// ---------------------------------------------------------------------------
// K3: per series b: propagate chunk-entry states: s_{c+1} = M^CHUNK * s_c + E_c
// M = [[-c1, -c0], [1, 0]] acting on (res_{t-1}, res_{t-2})
// ---------------------------------------------------------------------------
__global__ void __launch_bounds__(256)
k3_combine(const float* __restrict__ E, const float* __restrict__ ma_coef,
           float* __restrict__ Sst) {
  const int b = threadIdx.x;
  const float c0 = ma_coef[0], c1 = ma_coef[1];
  float m00 = -c1, m01 = -c0, m10 = 1.f, m11 = 0.f;
#pragma unroll
  for (int i = 0; i < 8; ++i) {        // M^256 by repeated squaring
    float n00 = m00*m00 + m01*m10, n01 = m00*m01 + m01*m11;
    float n10 = m10*m00 + m11*m10, n11 = m10*m01 + m11*m11;
    m00 = n00; m01 = n01; m10 = n10; m11 = n11;
  }
  float s1 = 0.f, s0 = 0.f;
  for (int c = 0; c < NCH; ++c) {
    Sst[(c * B_ + b) * 2 + 0] = s1;
    Sst[(c * B_ + b) * 2 + 1] = s0;
    float e1 = E[(c * B_ + b) * 2 + 0];
    float e0 = E[(c * B_ + b) * 2 + 1];
    float n1 = m00 * s1 + m01 * s0 + e1;
    float n0 = m10 * s1 + m11 * s0 + e0;
    s1 = n1; s0 = n0;
  }
}

// ---------------------------------------------------------------------------
// K4: per (c,b): rerun chunk with true entry state -> predT[t][b], chunk sums
// ---------------------------------------------------------------------------
__global__ void __launch_bounds__(256)
k4_phase3(const float* __restrict__ baseT, const float* __restrict__ dT,
          const float* __restrict__ Sst, const float* __restrict__ ma_coef,
          float* __restrict__ predT, float* __restrict__ Psum) {
  const int b = threadIdx.x, c = blockIdx.x;
  const float c0 = ma_coef[0], c1 = ma_coef[1];
  const int tb = c * CHUNK;
  const int te = (tb + CHUNK < STEPS) ? tb + CHUNK : STEPS;
  float s1 = Sst[(c * B_ + b) * 2 + 0];
  float s0 = Sst[(c * B_ + b) * 2 + 1];
  float psum = 0.f;
#pragma unroll 4
  for (int t = tb; t < te; ++t) {
    float mav  = (t >= 2) ? (c1 * s1 + c0 * s0) : 0.f;
    float res  = dT[(size_t)t * B_ + b] - mav;
    float pred = baseT[(size_t)t * B_ + b] + mav;
    predT[(size_t)t * B_ + b] = pred;
    psum += pred;
    s0 = s1; s1 = res;
  }
  Psum[c * B_ + b] = psum;
}

// ---------------------------------------------------------------------------
// K5: per series b: exclusive scan of chunk sums; write out[b][0] = x[b][0]
// ---------------------------------------------------------------------------
__global__ void __launch_bounds__(256)
k5_scan(const float* __restrict__ x, const float* __restrict__ Psum,
        float* __restrict__ Off, float* __restrict__ out) {
  const int b = threadIdx.x;
  out[(size_t)b * OUTW] = x[(size_t)b * T_];
  float run = 0.f;
  for (int c = 0; c < NCH; ++c) {
    Off[c * B_ + b] = run;
    run += Psum[c * B_ + b];
  }
}

// ---------------------------------------------------------------------------
// K6: per (c,b): out[b][t+1] = x[b][0] + Off + local running sum of pred
// ---------------------------------------------------------------------------
__global__ void __launch_bounds__(256)
k6_out(const float* __restrict__ x, const float* __restrict__ predT,
       const float* __restrict__ Off, float* __restrict__ out) {
  const int b = threadIdx.x, c = blockIdx.x;
  const int tb = c * CHUNK;
  const int te = (tb + CHUNK < STEPS) ? tb + CHUNK : STEPS;
  float run = x[(size_t)b * T_] + Off[c * B_ + b];
#pragma unroll 4
  for (int t = tb; t < te; ++t) {
    run += predT[(size_t)t * B_ + b];
    out[(size_t)b * OUTW + t + 1] = run;
  }
}

extern "C" void kernel_launch(void* const* d_in, const int* in_sizes, int n_in,
                              void* d_out, int out_size, void* d_ws, size_t ws_size,
                              hipStream_t stream) {
  const float* x        = (const float*)d_in[0];
  const float* features = (const float*)d_in[1];
  const float* ar_coef  = (const float*)d_in[2];
  const float* ma_coef  = (const float*)d_in[3];
  const float* fw       = (const float*)d_in[4];
  const float* bias     = (const float*)d_in[5];
  float* out = (float*)d_out;

  const size_t N1 = (size_t)STEPS * B_;
  float* baseT = (float*)d_ws;
  float* dT    = baseT + N1;
  float* predT = dT + N1;
  float* E     = predT + N1;           // NCH*B_*2
  float* Sst   = E + (size_t)NCH * B_ * 2;
  float* Psum  = Sst + (size_t)NCH * B_ * 2;
  float* Off   = Psum + (size_t)NCH * B_;

  // K1: 1024 16-timestep tiles per series, 8 waves (tiles) per block
  dim3 g1(128, B_);
  k1_base<<<g1, 256, 0, stream>>>(x, features, ar_coef, fw, bias, baseT, dT);
  k2_phase1<<<NCH, 256, 0, stream>>>(dT, ma_coef, E);
  k3_combine<<<1, 256, 0, stream>>>(E, ma_coef, Sst);
  k4_phase3<<<NCH, 256, 0, stream>>>(baseT, dT, Sst, ma_coef, predT, Psum);
  k5_scan<<<1, 256, 0, stream>>>(x, Psum, Off, out);
  k6_out<<<NCH, 256, 0, stream>>>(x, predT, Off, out);
}